// GCRN_19473381720231
// MI455X (gfx1250) — compile-verified
//
#include <hip/hip_runtime.h>
#include <hip/hip_bf16.h>
#include <math.h>

typedef __attribute__((ext_vector_type(2))) float v2f;
typedef __attribute__((ext_vector_type(8))) float v8f;

#define D_CH 32
#define G_CH 96   // 3 live gates (I, T, O) * 32 channels

// ---------------- normalization kernels ----------------

__global__ void gcrn_init_deg(float* deg, int n) {
    int i = blockIdx.x * blockDim.x + threadIdx.x;
    if (i < n) deg[i] = 1.0f;   // self-loop weight
}

__global__ void gcrn_acc_deg(const int* __restrict__ dst, const float* __restrict__ ew,
                             float* __restrict__ deg, int e) {
    int i = blockIdx.x * blockDim.x + threadIdx.x;
    if (i < e)
        __hip_atomic_fetch_add(&deg[dst[i]], ew[i], __ATOMIC_RELAXED, __HIP_MEMORY_SCOPE_AGENT);
}

__global__ void gcrn_fin_norm(float* deg, int n) {   // deg -> dinv in place
    int i = blockIdx.x * blockDim.x + threadIdx.x;
    if (i < n) {
        float d = deg[i];
        deg[i] = (d > 0.0f) ? rsqrtf(d) : 0.0f;
    }
}

__global__ void gcrn_edge_nw(const int* __restrict__ src, const int* __restrict__ dst,
                             const float* __restrict__ ew, const float* __restrict__ dinv,
                             float* __restrict__ nw, int e) {
    int i = blockIdx.x * blockDim.x + threadIdx.x;
    if (i < e) nw[i] = dinv[src[i]] * ew[i] * dinv[dst[i]];
}

// ---------------- dense transform: XW = x @ [W_I | W_T | W_O], fused self-loop init ----------------
// grid: N/16 blocks, 192 threads (6 waves). Wave t computes 16x16 tile of columns [16t,16t+16).
__global__ void gcrn_matmul_selfinit(const float* __restrict__ x,
                                     const float* __restrict__ Wl,   // conv_w + l*8*32*32
                                     const float* __restrict__ dinv,
                                     float* __restrict__ xw,         // [N,96]
                                     float* __restrict__ agg) {      // [N,96], init = dinv^2 * xw
    const int wave = threadIdx.x >> 5;          // 0..5
    const int lane = threadIdx.x & 31;
    const int n0   = lane & 15;
    const int hi   = lane >> 4;                 // 0: K pair {k,k+1} / M 0..7 ; 1: {k+2,k+3} / M 8..15
    const int row0 = blockIdx.x * 16;

    const int gate = wave >> 1;                 // 0,1,2 -> conv gate index 0,4,6
    const int g    = (gate == 0) ? 0 : (2 * gate + 2);
    const float* Wg = Wl + g * 32 * 32 + (wave & 1) * 16 + n0;   // column base
    const float* xr = x + (size_t)(row0 + n0) * D_CH;            // A-frag row

    v8f c = {};
    #pragma unroll
    for (int k = 0; k < 32; k += 4) {
        const int ka = k + 2 * hi;
        v2f a; a.x = xr[ka];            a.y = xr[ka + 1];
        v2f b; b.x = Wg[ka * 32];       b.y = Wg[(ka + 1) * 32];
        c = __builtin_amdgcn_wmma_f32_16x16x4_f32(false, a, false, b, (short)0, c, false, false);
    }

    const int col = wave * 16 + n0;
    #pragma unroll
    for (int j = 0; j < 8; ++j) {
        const int r = row0 + hi * 8 + j;
        const float v  = c[j];
        const float dv = dinv[r];
        xw [(size_t)r * G_CH + col] = v;
        agg[(size_t)r * G_CH + col] = dv * dv * v;   // self-loop term of gcn_norm
    }
}

// ---------------- edge scatter-add: one wave per edge, 96 channels coalesced ----------------
__global__ void gcrn_edge_agg(const int* __restrict__ src, const int* __restrict__ dst,
                              const float* __restrict__ nw,
                              const float* __restrict__ xw, float* __restrict__ agg, int e) {
    const int edge = blockIdx.x * (blockDim.x >> 5) + (threadIdx.x >> 5);
    if (edge >= e) return;
    const int lane = threadIdx.x & 31;
    const int s = src[edge], d = dst[edge];
    const float w = nw[edge];
    const float* xs = xw  + (size_t)s * G_CH;
    float*       ad = agg + (size_t)d * G_CH;
    #pragma unroll
    for (int j = 0; j < 3; ++j) {
        const int cc = lane + 32 * j;
        __hip_atomic_fetch_add(&ad[cc], w * xs[cc], __ATOMIC_RELAXED, __HIP_MEMORY_SCOPE_AGENT);
    }
}

// ---------------- fused gates (F is dead: C_old == 0) ----------------
__global__ void gcrn_gates(const float* __restrict__ agg,
                           const float* __restrict__ cb,   // conv_b + l*8*32
                           const float* __restrict__ bgl,  // bg + l*4*32
                           const float* __restrict__ wco,  // wc + (l*3+2)*32
                           float* __restrict__ out, int n) {
    const int idx = blockIdx.x * blockDim.x + threadIdx.x;
    if (idx >= n * D_CH) return;
    const int c    = idx & 31;
    const int node = idx >> 5;
    const float* a = agg + (size_t)node * G_CH;

    const float bi = cb[c]          + cb[32 + c]      + bgl[c];
    const float bt = cb[4 * 32 + c] + cb[5 * 32 + c]  + bgl[2 * 32 + c];
    const float bo = cb[6 * 32 + c] + cb[7 * 32 + c]  + bgl[3 * 32 + c];

    const float I = 1.0f / (1.0f + expf(-(a[c] + bi)));
    const float T = tanhf(a[32 + c] + bt);
    const float C = I * T;
    const float O = 1.0f / (1.0f + expf(-(a[64 + c] + bo + wco[c] * C)));
    out[idx] = O * tanhf(C);
}

// ---------------- host launcher ----------------
extern "C" void kernel_launch(void* const* d_in, const int* in_sizes, int n_in,
                              void* d_out, int out_size, void* d_ws, size_t ws_size,
                              hipStream_t stream) {
    const float* X       = (const float*)d_in[0];
    const int*   eidx    = (const int*)  d_in[1];
    const float* ew      = (const float*)d_in[2];
    const float* conv_w  = (const float*)d_in[3];
    const float* conv_b  = (const float*)d_in[4];
    const float* wc      = (const float*)d_in[5];
    const float* bg      = (const float*)d_in[6];

    const int N = in_sizes[0] / D_CH;
    const int E = in_sizes[2];
    const int L = in_sizes[3] / (8 * D_CH * D_CH);
    const int* src = eidx;
    const int* dst = eidx + E;

    float* ws   = (float*)d_ws;
    float* dinv = ws;                               // N   (deg -> dinv in place)
    float* nw   = dinv + N;                         // E
    float* xw   = nw   + E;                         // N*96
    float* agg  = xw   + (size_t)N * G_CH;          // N*96
    float* h1   = agg  + (size_t)N * G_CH;          // N*32
    float* h2   = h1   + (size_t)N * D_CH;          // N*32

    const int TB = 256;
    // normalization (once)
    gcrn_init_deg<<<(N + TB - 1) / TB, TB, 0, stream>>>(dinv, N);
    gcrn_acc_deg <<<(E + TB - 1) / TB, TB, 0, stream>>>(dst, ew, dinv, E);
    gcrn_fin_norm<<<(N + TB - 1) / TB, TB, 0, stream>>>(dinv, N);
    gcrn_edge_nw <<<(E + TB - 1) / TB, TB, 0, stream>>>(src, dst, ew, dinv, nw, E);

    const float* xin = X;
    for (int l = 0; l < L; ++l) {
        gcrn_matmul_selfinit<<<N / 16, 192, 0, stream>>>(
            xin, conv_w + (size_t)l * 8 * 32 * 32, dinv, xw, agg);

        const int edgesPerBlock = TB / 32;
        gcrn_edge_agg<<<(E + edgesPerBlock - 1) / edgesPerBlock, TB, 0, stream>>>(
            src, dst, nw, xw, agg, E);

        float* out_l = (l == L - 1) ? (float*)d_out : ((l & 1) ? h2 : h1);
        gcrn_gates<<<((size_t)N * D_CH + TB - 1) / TB, TB, 0, stream>>>(
            agg, conv_b + (size_t)l * 8 * 32, bg + (size_t)l * 4 * 32,
            wc + ((size_t)l * 3 + 2) * 32, out_l, N);
        xin = out_l;
    }
}